// ctc_loss_53850299957539
// MI455X (gfx1250) — compile-verified
//
#include <hip/hip_runtime.h>
#include <stdint.h>

// CTC forward loss for MI455X (gfx1250).
// B blocks (one per batch), BS threads. Alpha recurrence in LDS (double buffer),
// emission tiles streamed into LDS with the Tensor Data Mover (tensor_load_to_lds)
// and synchronized with s_wait_tensorcnt. wave32-aware (wave 0 drives the TDM).
// Time loop clamped to Teff = min(T, output_length): frozen steps are identities.

#define BS      128          // threads per block (4 waves)
#define CCLS    96           // number of classes (C)
#define SMAX    401          // max extended path length (2*200+1)
#define NS      4            // ceil(SMAX / BS)
#define TILE_T  64           // timesteps per LDS tile (64*96*4 = 24 KB per buffer)
#define NEGV    (-1e30f)

#if defined(__has_builtin)
#if __has_builtin(__builtin_amdgcn_tensor_load_to_lds)
#define HAS_TDM 1
#endif
#endif
#ifndef HAS_TDM
#define HAS_TDM 0
#endif

typedef unsigned int u32x4 __attribute__((ext_vector_type(4)));
typedef int          i32x8 __attribute__((ext_vector_type(8)));
typedef int          i32x4 __attribute__((ext_vector_type(4)));

// logaddexp via hardware transcendentals (v_exp_f32 / v_log_f32, base-2).
// Safe for NEGV sentinels: exp2(-huge) -> 0 -> returns max(a,b).
__device__ __forceinline__ float lae(float a, float b) {
    float m = fmaxf(a, b);
    float d = fminf(a, b) - m;                       // <= 0
    float z = __builtin_amdgcn_exp2f(d * 1.4426950408889634f);
    return m + 0.6931471805599453f * __builtin_amdgcn_logf(1.0f + z);
}

#if HAS_TDM
// Issue a 2-D TDM load: rows x CCLS fp32 tile, row stride CCLS elements.
// D# layout per CDNA5 ISA ch.8 (group0: count/lds_addr/global_addr/type,
// group1: data_size/tensor dims/tile dims/strides). Groups 2,3 zero (2-D tensor).
__device__ __forceinline__ void issue_tdm_2d(const float* gsrc, unsigned lds_off,
                                             int rows, int rows_remaining) {
    unsigned long long ga = (unsigned long long)gsrc;
    u32x4 g0;
    g0.x = 1u;                                            // count=1 (valid user desc)
    g0.y = lds_off;                                       // lds_addr (bytes)
    g0.z = (unsigned)(ga & 0xFFFFFFFFull);                // global_addr[31:0]
    g0.w = ((unsigned)(ga >> 32) & 0x01FFFFFFu)           // global_addr[56:32]
         | 0x80000000u;                                   // type=2 ("image")
    unsigned td0 = CCLS;                                  // tensor_dim0 (elements)
    unsigned td1 = (unsigned)rows_remaining;              // tensor_dim1 (rows left, OOB clamp)
    unsigned st0 = CCLS;                                  // tensor_dim0_stride (elements)
    i32x8 g1;
    g1[0] = (int)(2u << 16);                              // data_size=2 -> 4 bytes
    g1[1] = (int)((td0 & 0xFFFFu) << 16);                 // tensor_dim0[15:0] @ bits 63:48
    g1[2] = (int)(((td1 & 0xFFFFu) << 16) | (td0 >> 16)); // dim1 lo | dim0 hi
    g1[3] = (int)(((unsigned)CCLS << 16)                  // tile_dim0=96 @ bits 127:112
                  | ((td1 >> 16) & 0xFFFFu));             // tensor_dim1[31:16]
    g1[4] = (int)((unsigned)rows & 0xFFFFu);              // tile_dim1 = rows (tile_dim2=0)
    g1[5] = (int)st0;                                     // tensor_dim0_stride[31:0]
    g1[6] = 0;                                            // stride hi | dim1_stride lo
    g1[7] = 0;
    i32x4 z4 = {0, 0, 0, 0};
    i32x8 z8 = {0, 0, 0, 0, 0, 0, 0, 0};
    // 6-arg form (clang-23 / therock-10.0 headers): groups 0..3 + extra group + cpol
    __builtin_amdgcn_tensor_load_to_lds(g0, g1, z4, z4, z8, 0);
}
#endif

__device__ __forceinline__ void tile_issue(const float* gsrc, float* ldst,
                                           int rows, int rows_remaining, int tid) {
#if HAS_TDM
    if (tid < 32) {  // wave 0 drives the TDM (EXEC-independent DMA)
        issue_tdm_2d(gsrc, (unsigned)(unsigned long long)(void*)ldst, rows, rows_remaining);
    }
#else
    int n = rows * CCLS;
    for (int i = tid; i < n; i += BS) ldst[i] = gsrc[i];
#endif
}

__device__ __forceinline__ void tile_wait(int tid) {
#if HAS_TDM
    if (tid < 32) __builtin_amdgcn_s_wait_tensorcnt(0);
#endif
}

__global__ __launch_bounds__(BS)
void ctc_alpha_kernel(const float* __restrict__ outputs,   // [B, T, C] log-probs
                      const int*   __restrict__ labels,    // [B, L]
                      const int*   __restrict__ out_lens,  // [B]
                      const int*   __restrict__ lab_lens,  // [B]
                      float* __restrict__ per_batch,       // [B] nll / label_len
                      int T, int L) {
    __shared__ float tile[2][TILE_T * CCLS];   // 48 KB emission double buffer
    __shared__ float alpha[2][SMAX + 2];       // [0..1] = NEG pad, s maps to s+2

    const int b      = blockIdx.x;
    const int tid    = threadIdx.x;
    const int outlen = out_lens[b];
    const int lablen = lab_lens[b];
    const int Srun   = 2 * L + 1;
    const int Teff   = (outlen < T) ? outlen : T;  // frozen steps are identities

    // Per-thread owned extended-path positions: class id + skip-allow flag.
    int  extc[NS];
    bool allow[NS];
    #pragma unroll
    for (int j = 0; j < NS; ++j) {
        int s = tid + j * BS;
        int c = 0; bool al = false;
        if (s < Srun && (s & 1)) {
            int li = s >> 1;
            c = labels[b * L + li];
            if (s >= 3) al = (c != labels[b * L + li - 1]);
        }
        extc[j] = c; allow[j] = al;
    }

    // Init both alpha buffers (incl. the 2-element NEG pad).
    for (int i = tid; i < SMAX + 2; i += BS) {
        alpha[0][i] = NEGV;
        alpha[1][i] = NEGV;
    }

    const int ntiles = (Teff + TILE_T - 1) / TILE_T;
    const float* gbase = outputs + (long long)b * T * CCLS;

    // Prime the pipeline: tile 0.
    tile_issue(gbase, &tile[0][0], (Teff < TILE_T ? Teff : TILE_T), Teff, tid);

    int pc = 0;  // alpha buffer parity holding the current alpha
    for (int ti = 0; ti < ntiles; ++ti) {
        const int t0   = ti * TILE_T;
        const int rows = (Teff - t0 < TILE_T) ? (Teff - t0) : TILE_T;

        tile_wait(tid);        // wave0: current tile DMA complete
        __syncthreads();       // everyone: tile + alpha-init visible

        // Prefetch next tile into the other buffer while we compute.
        const int t0n = t0 + TILE_T;
        if (t0n < Teff) {
            const int rn = (Teff - t0n < TILE_T) ? (Teff - t0n) : TILE_T;
            tile_issue(gbase + (long long)t0n * CCLS, &tile[(ti + 1) & 1][0],
                       rn, Teff - t0n, tid);
        }

        const float* __restrict__ em = &tile[ti & 1][0];
        int ttStart = 0;

        if (ti == 0) {
            // t = 0 initialization: alpha0[0] = emit[0,0], alpha0[1] = emit[0,1].
            float* a0 = &alpha[pc][0];
            #pragma unroll
            for (int j = 0; j < NS; ++j) {
                int s = tid + j * BS;
                if (s < Srun) a0[s + 2] = (s < 2) ? em[extc[j]] : NEGV;
            }
            __syncthreads();
            ttStart = 1;
        }

        // Branch-free hot loop: every t in [ttStart, rows) is an active step.
        for (int tt = ttStart; tt < rows; ++tt) {
            const float* __restrict__ erow = em + tt * CCLS;
            const float* __restrict__ acur = &alpha[pc][0];
            float*       __restrict__ anxt = &alpha[pc ^ 1][0];

            float nv[NS];
            #pragma unroll
            for (int j = 0; j < NS; ++j) {
                int s = tid + j * BS;
                if (s < Srun) {
                    float a  = acur[s + 2];
                    float a1 = acur[s + 1];
                    float a2 = allow[j] ? acur[s] : NEGV;
                    nv[j] = lae(lae(a, a1), a2) + erow[extc[j]];
                }
            }
            #pragma unroll
            for (int j = 0; j < NS; ++j) {
                int s = tid + j * BS;
                if (s < Srun) anxt[s + 2] = nv[j];
            }
            __syncthreads();                  // single barrier per timestep
            pc ^= 1;
        }
    }

    if (tid == 0) {
        const float* af = &alpha[pc][0];
        int end = 2 * lablen;
        float nll = -lae(af[end + 2], af[end + 1]);
        per_batch[b] = nll / (float)lablen;
    }
}

__global__ void ctc_reduce_kernel(const float* __restrict__ per_batch,
                                  float* __restrict__ out, int B) {
    if (threadIdx.x == 0 && blockIdx.x == 0) {
        float s = 0.0f;
        for (int i = 0; i < B; ++i) s += per_batch[i];
        out[0] = s / (float)B;
    }
}

extern "C" void kernel_launch(void* const* d_in, const int* in_sizes, int n_in,
                              void* d_out, int out_size, void* d_ws, size_t ws_size,
                              hipStream_t stream) {
    const float* outputs = (const float*)d_in[0];   // [B,T,C] fp32 log-probs
    const int*   labels  = (const int*)d_in[1];     // [B,L]
    const int*   olen    = (const int*)d_in[2];     // [B]
    const int*   llen    = (const int*)d_in[3];     // [B]

    const int B = in_sizes[2];
    const int L = in_sizes[1] / B;
    const int T = in_sizes[0] / (B * CCLS);

    float* perb = (float*)d_ws;                     // B floats of scratch

    ctc_alpha_kernel<<<B, BS, 0, stream>>>(outputs, labels, olen, llen, perb, T, L);
    ctc_reduce_kernel<<<1, 32, 0, stream>>>(perb, (float*)d_out, B);
}